// ProjectiveSpatialTransformer_73220602462351
// MI455X (gfx1250) — compile-verified
//
#include <hip/hip_runtime.h>

// Trilinear forward-splat (scatter-add), H=W=D=192, fp32.
// Memory-bound: ~142 MB traffic -> ~6us floor at 23.3 TB/s.
// Strategy: NT-hint streaming loads for src/flow (read-once data must not
// evict the 28MB output, which lives in the 192MB L2), b128 vector loads
// (4 voxels/thread along contiguous D), non-returning global_atomic_add_f32.

#define H_  192
#define W_  192
#define D_  192
#define WD_ (W_ * D_)          // 36864
#define N_  (H_ * W_ * D_)     // 7077888

typedef float float4v __attribute__((ext_vector_type(4)));

__global__ __launch_bounds__(256) void splat_zero_kernel(float* __restrict__ out) {
    int t = blockIdx.x * blockDim.x + threadIdx.x;   // one float4 per thread
    float4v z = {0.0f, 0.0f, 0.0f, 0.0f};
    ((float4v*)out)[t] = z;                          // global_store_b128
}

__global__ __launch_bounds__(256) void splat_scatter_kernel(
        const float* __restrict__ src,
        const float* __restrict__ flow,
        float* __restrict__ out) {
    int t = blockIdx.x * blockDim.x + threadIdx.x;
    int n = t * 4;                       // base voxel; 4 consecutive d-voxels

    // Decompose (i, j, k0): groups of 4 never cross a row since 192 % 4 == 0.
    int i   = n / WD_;
    int rem = n - i * WD_;
    int j   = rem / D_;
    int k0  = rem - j * D_;

    // Streaming, read-once inputs: non-temporal b128 loads (CDNA5 TH=NT)
    // so they don't evict the L2-resident output volume.
    float4v s4  = __builtin_nontemporal_load((const float4v*)(src  + n));
    float4v fx4 = __builtin_nontemporal_load((const float4v*)(flow + n));
    float4v fy4 = __builtin_nontemporal_load((const float4v*)(flow + N_ + n));
    float4v fz4 = __builtin_nontemporal_load((const float4v*)(flow + 2 * N_ + n));

    const float fi = (float)i;
    const float fj = (float)j;

#pragma unroll
    for (int v = 0; v < 4; ++v) {
        float s  = s4[v];
        // locs = grid + flow, clamped below at 0 (reference: where(locs>0, locs, 0))
        float lx = fmaxf(fi + fx4[v], 0.0f);
        float ly = fmaxf(fj + fy4[v], 0.0f);
        float lz = fmaxf((float)(k0 + v) + fz4[v], 0.0f);

        float flx = floorf(lx), fly = floorf(ly), flz = floorf(lz);
        float dx = lx - flx, dy = ly - fly, dz = lz - flz;

        // base = min(int(floor(locs)), dim-1)
        int bx = min((int)flx, H_ - 1);
        int by = min((int)fly, W_ - 1);
        int bz = min((int)flz, D_ - 1);
        // corner = min(base + 1, dim-1)
        int bx1 = min(bx + 1, H_ - 1);
        int by1 = min(by + 1, W_ - 1);
        int bz1 = min(bz + 1, D_ - 1);

        // Fold src into the x-weights: 14 muls total for the 8 corner weights.
        float ax0 = s - s * dx;          // s*(1-dx)  (fma)
        float ax1 = s * dx;
        float w00 = ax0 * (1.0f - dy);
        float w01 = ax0 * dy;
        float w10 = ax1 * (1.0f - dy);
        float w11 = ax1 * dy;
        float wz0 = 1.0f - dz;

        int r00 = bx  * WD_ + by  * D_;
        int r01 = bx  * WD_ + by1 * D_;
        int r10 = bx1 * WD_ + by  * D_;
        int r11 = bx1 * WD_ + by1 * D_;

        // 8 non-returning global_atomic_add_f32; targets are near-local
        // (flow ~ N(0,1)) and the whole 28MB output fits in L2.
        atomicAdd(out + r00 + bz,  w00 * wz0);
        atomicAdd(out + r00 + bz1, w00 * dz);
        atomicAdd(out + r01 + bz,  w01 * wz0);
        atomicAdd(out + r01 + bz1, w01 * dz);
        atomicAdd(out + r10 + bz,  w10 * wz0);
        atomicAdd(out + r10 + bz1, w10 * dz);
        atomicAdd(out + r11 + bz,  w11 * wz0);
        atomicAdd(out + r11 + bz1, w11 * dz);
    }
}

extern "C" void kernel_launch(void* const* d_in, const int* in_sizes, int n_in,
                              void* d_out, int out_size, void* d_ws, size_t ws_size,
                              hipStream_t stream) {
    const float* src  = (const float*)d_in[0];   // [1,1,192,192,192] f32
    const float* flow = (const float*)d_in[1];   // [1,3,192,192,192] f32
    float* out = (float*)d_out;                  // [1,1,192,192,192] f32

    // N_/(256 threads * 4 floats) = 6912 blocks for both kernels.
    const int blocks = N_ / (256 * 4);

    // Output must be re-zeroed every launch (harness poisons d_out once and
    // replays the captured graph; scatter-add accumulates).
    splat_zero_kernel<<<blocks, 256, 0, stream>>>(out);
    splat_scatter_kernel<<<blocks, 256, 0, stream>>>(src, flow, out);
}